// get_one_hot_59442347376951
// MI455X (gfx1250) — compile-verified
//
#include <hip/hip_runtime.h>

// One-hot expansion: label (4096x4096 int32, 0..7) -> out (8, 4096, 4096) f32.
// Pure streaming kernel: 64 MB read + 512 MB write => ~25 us floor @ 23.3 TB/s.
// Strategy: b128 loads/stores, non-temporal hints (write-once data >> 192 MB L2),
// global_prefetch_b8 for the label stream, grid-stride x2 per thread.

typedef int   v4i __attribute__((ext_vector_type(4)));
typedef float v4f __attribute__((ext_vector_type(4)));

namespace {
constexpr int       kN   = 8;
constexpr long long kHW  = 4096LL * 4096LL;        // 16,777,216 pixels per plane
constexpr int       kHW4 = (int)(kHW / 4);         // 4,194,304 int4 chunks
constexpr int       kBlock = 256;                  // 8 wave32s
constexpr int       kGrid  = 8192;                 // -> exactly 2 iterations/thread
}

__global__ __launch_bounds__(kBlock)
void get_one_hot_59442347376951_kernel(const v4i* __restrict__ lab4,
                                       float* __restrict__ out) {
  const int stride = gridDim.x * blockDim.x;
  for (int i = blockIdx.x * blockDim.x + threadIdx.x; i < kHW4; i += stride) {
    // Prefetch next label chunk for this thread (gfx1250: global_prefetch_b8).
    if (i + stride < kHW4) {
      __builtin_prefetch(&lab4[i + stride], /*rw=*/0, /*locality=*/0);
    }

    // Read 4 labels as one b128, non-temporal (read-once stream).
    const v4i L = __builtin_nontemporal_load(&lab4[i]);

    // Emit one b128 NT store per output plane; consecutive threads write
    // consecutive 16-B chunks within each plane -> fully coalesced.
#pragma unroll
    for (int n = 0; n < kN; ++n) {
      v4f v;
      v.x = (L.x == n) ? 1.0f : 0.0f;
      v.y = (L.y == n) ? 1.0f : 0.0f;
      v.z = (L.z == n) ? 1.0f : 0.0f;
      v.w = (L.w == n) ? 1.0f : 0.0f;
      v4f* dst = (v4f*)(out + (long long)n * kHW) + i;
      __builtin_nontemporal_store(v, dst);
    }
  }
}

extern "C" void kernel_launch(void* const* d_in, const int* in_sizes, int n_in,
                              void* d_out, int out_size, void* d_ws, size_t ws_size,
                              hipStream_t stream) {
  // d_in[0] = label (4096*4096 int32), d_in[1] = N (scalar, known == 8).
  const v4i* lab4 = (const v4i*)d_in[0];
  float* out = (float*)d_out;

  get_one_hot_59442347376951_kernel<<<dim3(kGrid), dim3(kBlock), 0, stream>>>(lab4, out);
}